// fofe_encoder_1374389534941
// MI455X (gfx1250) — compile-verified
//
#include <hip/hip_runtime.h>

typedef __attribute__((ext_vector_type(2))) float v2f;
typedef __attribute__((ext_vector_type(8))) float v8f;

// Problem constants (match reference setup_inputs)
#define B_  8
#define C_  256
#define T_  512
#define L_  16
#define TP1 513      // T+1
#define NT  33       // ceil(513/16) t-tiles

static __device__ __forceinline__ int clampi(int v, int lo, int hi) {
    return v < lo ? lo : (v > hi ? hi : v);
}

// D = coeff(16xK=16) x S(16x 16-col tile) via 4 chained V_WMMA_F32_16X16X4_F32.
// One wave per (b,c, 16-column tile); 4 output matrices per wave.
__global__ __launch_bounds__(128) void fofe_wmma_kernel(
    const float* __restrict__ x,
    const float* __restrict__ a_lf,
    const float* __restrict__ a_hf,
    const float* __restrict__ a_li,
    const float* __restrict__ a_hi,
    float* __restrict__ out)
{
    const int ln   = threadIdx.x;           // lane 0..31
    const int wave = threadIdx.y;           // 0..3
    const int pair = blockIdx.y * 4 + wave; // 0..2047  (b*C + c)
    const int b    = pair >> 8;
    const int c    = pair & (C_ - 1);

    // Last tile overlaps backward so every lane's column is in-range:
    // keeps EXEC all-ones (WMMA requirement); overlap re-stores identical values.
    const int t0  = (blockIdx.x == NT - 1) ? (TP1 - 16) : (blockIdx.x << 4);
    const int lo  = ln & 15;                // A-row l, B/D column within tile
    const int g   = ln >> 4;                // lane group (K / row-half select)
    const int t   = t0 + lo;                // global output column, 0..512

    const float* xrow = x + (size_t)pair * T_;

    // Per-lane alphas for this lane's coefficient row l = lo
    const float alf = a_lf[lo], ahf = a_hf[lo];
    const float ali = a_li[lo], ahi = a_hi[lo];

    const float alf2 = alf * alf, alf4 = alf2 * alf2;
    const float ahf2 = ahf * ahf, ahf4 = ahf2 * ahf2;
    const float ali2 = ali * ali, ali4 = ali2 * ali2;
    const float ahi2 = ahi * ahi, ahi4 = ahi2 * ahi2;

    float plf = 1.f, phf = 1.f, pli = 1.f, phi = 1.f; // alpha^(4j)

    v8f acc_fl = {0.f,0.f,0.f,0.f,0.f,0.f,0.f,0.f};
    v8f acc_fh = acc_fl, acc_il = acc_fl, acc_ih = acc_fl;

#pragma unroll
    for (int j = 0; j < 4; ++j) {
        const int m0 = 4 * j + 2 * g;       // this lane's first K index in chunk j

        // ---- B fragments: branchless edge handling ----
        // Clamp index into [0, T-1] -> unconditional load, then select 0.
        v2f bf;  // fwd: S_fwd[m][t] = x[t-(m+1)], zero for t-(m+1) < 0
        {
            const int i0 = t - (m0 + 1);
            const int i1 = t - (m0 + 2);
            const float v0 = xrow[clampi(i0, 0, T_ - 1)];
            const float v1 = xrow[clampi(i1, 0, T_ - 1)];
            bf.x = (i0 >= 0) ? v0 : 0.f;
            bf.y = (i1 >= 0) ? v1 : 0.f;
        }
        v2f bi;  // inv: S_inv[k][t] = x[t+k], zero for t+k >= T
        {
            const int i0 = t + m0;
            const int i1 = i0 + 1;
            const float v0 = xrow[clampi(i0, 0, T_ - 1)];
            const float v1 = xrow[clampi(i1, 0, T_ - 1)];
            bi.x = (i0 < T_) ? v0 : 0.f;
            bi.y = (i1 < T_) ? v1 : 0.f;
        }

        // ---- A fragments: coeff[l][m] = (m<=l) ? alpha_l^m : 0 ----
        v2f afl, afh, ail, aih;
        {
            const float p0 = g ? plf * alf2 : plf;
            afl.x = (m0     <= lo) ? p0       : 0.f;
            afl.y = (m0 + 1 <= lo) ? p0 * alf : 0.f;
        }
        {
            const float p0 = g ? phf * ahf2 : phf;
            afh.x = (m0     <= lo) ? p0       : 0.f;
            afh.y = (m0 + 1 <= lo) ? p0 * ahf : 0.f;
        }
        {
            const float p0 = g ? pli * ali2 : pli;
            ail.x = (m0     <= lo) ? p0       : 0.f;
            ail.y = (m0 + 1 <= lo) ? p0 * ali : 0.f;
        }
        {
            const float p0 = g ? phi * ahi2 : phi;
            aih.x = (m0     <= lo) ? p0       : 0.f;
            aih.y = (m0 + 1 <= lo) ? p0 * ahi : 0.f;
        }
        plf *= alf4; phf *= ahf4; pli *= ali4; phi *= ahi4;

        // ---- K-chunk accumulate: D = A*B + C ----
        acc_fl = __builtin_amdgcn_wmma_f32_16x16x4_f32(false, afl, false, bf,
                                                       (short)0, acc_fl, false, false);
        acc_fh = __builtin_amdgcn_wmma_f32_16x16x4_f32(false, afh, false, bf,
                                                       (short)0, acc_fh, false, false);
        acc_il = __builtin_amdgcn_wmma_f32_16x16x4_f32(false, ail, false, bi,
                                                       (short)0, acc_il, false, false);
        acc_ih = __builtin_amdgcn_wmma_f32_16x16x4_f32(false, aih, false, bi,
                                                       (short)0, acc_ih, false, false);
    }

    // ---- stores: out = fwd [B,2C,L,513] then inv [B,2C,L,513] ----
    // Streaming 269MB with zero reuse -> non-temporal to spare L2 for x.
    const size_t half   = (size_t)B_ * (2 * C_) * L_ * TP1;  // fwd output elems
    const size_t base_l = ((size_t)(b * (2 * C_) + c)      * L_) * TP1;
    const size_t base_h = ((size_t)(b * (2 * C_) + c + C_) * L_) * TP1;

#pragma unroll
    for (int r = 0; r < 8; ++r) {
        const int lrow = r + 8 * g;                 // D row for this lane/VGPR
        const size_t off = (size_t)lrow * TP1 + t;  // contiguous in t across lanes 0..15
        __builtin_nontemporal_store(acc_fl[r], out + base_l + off);
        __builtin_nontemporal_store(acc_fh[r], out + base_h + off);
        __builtin_nontemporal_store(acc_il[r], out + half + base_l + off);
        __builtin_nontemporal_store(acc_ih[r], out + half + base_h + off);
    }
}

extern "C" void kernel_launch(void* const* d_in, const int* in_sizes, int n_in,
                              void* d_out, int out_size, void* d_ws, size_t ws_size,
                              hipStream_t stream) {
    const float* x    = (const float*)d_in[0];
    const float* a_lf = (const float*)d_in[1];
    const float* a_hf = (const float*)d_in[2];
    const float* a_li = (const float*)d_in[3];
    const float* a_hi = (const float*)d_in[4];
    float* out = (float*)d_out;

    dim3 block(32, 4, 1);            // 4 wave32s per block
    dim3 grid(NT, (B_ * C_) / 4, 1); // 33 t-tiles x 512 pair-groups
    fofe_wmma_kernel<<<grid, block, 0, stream>>>(x, a_lf, a_hf, a_li, a_hi, out);
}